// EdgeWeightedGCNLayer_91242285236401
// MI455X (gfx1250) — compile-verified
//
#include <hip/hip_runtime.h>
#include <hip/hip_bf16.h>

typedef __attribute__((ext_vector_type(2))) float v2f;
typedef __attribute__((ext_vector_type(8))) float v8f;

#define IN_DIM 128
#define OUT_DIM 128

// -----------------------------------------------------------------------------
// Kernel 1: h[node, o] = sum_i x[node, i] * W[o, i] + b[o]
// One wave (32 lanes) computes a 16-node x 128-output stripe via
// V_WMMA_F32_16X16X4_F32: 8 C tiles (16x16), K=128 -> 32 WMMA steps each.
//
// Fragment layouts per CDNA5 ISA 7.12.2 (wave32):
//   A (16x4, M x K):  lanes 0-15 hold M=lane, VGPR0=K0, VGPR1=K1 (K0,K1 = 0,1)
//                     lanes 16-31 hold M=lane-16, K0,K1 = 2,3
//   B (4x16, K x N):  row K striped across lanes within a VGPR, mirrored halves:
//                     VGPR0: lanes0-15 K=0, lanes16-31 K=2; VGPR1: K=1 / K=3
//   C/D (16x16 f32):  VGPR v: lanes0-15 -> M=v, N=lane; lanes16-31 -> M=v+8
// -----------------------------------------------------------------------------
__global__ void __launch_bounds__(128)
gcn_linear_wmma(const float* __restrict__ x, const float* __restrict__ W,
                const float* __restrict__ bias, float* __restrict__ h,
                int totalNodes) {
  const int lane = threadIdx.x & 31;
  const int wave = threadIdx.x >> 5;
  const int tile = blockIdx.x * (blockDim.x >> 5) + wave;
  const int nodeBase = tile * 16;
  if (nodeBase >= totalNodes) return;

  const int half = lane >> 4;   // 0: lanes 0-15, 1: lanes 16-31
  const int l    = lane & 15;   // row (A) / column (B,C,D) index
  const int kA0  = half * 2;    // this lane's K offsets within a K=4 step
  const int kA1  = kA0 + 1;

  v8f acc[8] = {};              // 8 output tiles of 16 columns -> 128 outputs

  const float* __restrict__ xrow = x + (size_t)(nodeBase + l) * IN_DIM;

  for (int k = 0; k < IN_DIM; k += 4) {
    v2f a;
    a.x = xrow[k + kA0];
    a.y = xrow[k + kA1];
#pragma unroll
    for (int t = 0; t < 8; ++t) {
      const int n = t * 16 + l;               // output channel for this lane
      const float* __restrict__ wrow = W + (size_t)n * IN_DIM;
      v2f bfrag;
      bfrag.x = wrow[k + kA0];                // B[K, N=n] = W[n, K]
      bfrag.y = wrow[k + kA1];
      acc[t] = __builtin_amdgcn_wmma_f32_16x16x4_f32(
          /*neg_a=*/false, a, /*neg_b=*/false, bfrag,
          /*c_mod=*/(short)0, acc[t], /*reuse_a=*/false, /*reuse_b=*/false);
    }
  }

#pragma unroll
  for (int t = 0; t < 8; ++t) {
    const int n = t * 16 + l;
    const float bv = bias[n];
#pragma unroll
    for (int v = 0; v < 8; ++v) {
      const int m = v + half * 8;             // C/D row held by this lane/VGPR
      h[(size_t)(nodeBase + m) * OUT_DIM + n] = acc[t][v] + bv;
    }
  }
}

// -----------------------------------------------------------------------------
// Kernel 2: zero the output (harness poisons d_out; must re-zero every call).
// -----------------------------------------------------------------------------
__global__ void zero_out(float* __restrict__ out, long long n) {
  long long i = (long long)blockIdx.x * blockDim.x + threadIdx.x;
  if (i < n) out[i] = 0.0f;
}

// -----------------------------------------------------------------------------
// Kernel 3: out[b, tgt[e], c] += ew[b, e] * h[b, src[e], c]
// 128 consecutive threads = 128 channels of one edge -> src/tgt/w are
// wave-uniform, channel accesses fully coalesced, f32 atomics resolve in L2
// (entire out tensor ~41 MB fits the 192 MB L2).
// -----------------------------------------------------------------------------
__global__ void __launch_bounds__(256)
edge_scatter(const float* __restrict__ h, const int* __restrict__ srcIdx,
             const int* __restrict__ tgtIdx, const float* __restrict__ ew,
             float* __restrict__ out, int E, int N, int B) {
  long long idx = (long long)blockIdx.x * blockDim.x + threadIdx.x;
  long long total = (long long)E * OUT_DIM;
  if (idx >= total) return;
  const int c = (int)(idx & (OUT_DIM - 1));
  const int e = (int)(idx >> 7);
  const int s = srcIdx[e];
  const int t = tgtIdx[e];
  for (int b = 0; b < B; ++b) {
    const float w = ew[(size_t)b * E + e];
    const float val = w * h[((size_t)b * N + s) * OUT_DIM + c];
    atomicAdd(out + ((size_t)b * N + t) * OUT_DIM + c, val);
  }
}

extern "C" void kernel_launch(void* const* d_in, const int* in_sizes, int n_in,
                              void* d_out, int out_size, void* d_ws, size_t ws_size,
                              hipStream_t stream) {
  const float* x    = (const float*)d_in[0];   // [B, N, 128]
  const int*   ei   = (const int*)d_in[1];     // [2, E]
  const float* ew   = (const float*)d_in[2];   // [B, E]
  const float* W    = (const float*)d_in[3];   // [128, 128]
  const float* bias = (const float*)d_in[4];   // [128]

  const int E = in_sizes[1] / 2;
  const int B = in_sizes[2] / E;
  const int totalNodes = in_sizes[0] / IN_DIM; // B * N
  const int N = totalNodes / B;

  float* h   = (float*)d_ws;                   // [B*N, 128] scratch (~41 MB)
  float* out = (float*)d_out;

  // 1) Linear transform via WMMA: 4 waves / block, one 16-node tile per wave.
  {
    const int tiles = (totalNodes + 15) / 16;
    const int wavesPerBlock = 4;
    const int blocks = (tiles + wavesPerBlock - 1) / wavesPerBlock;
    gcn_linear_wmma<<<blocks, wavesPerBlock * 32, 0, stream>>>(x, W, bias, h,
                                                               totalNodes);
  }

  // 2) Zero output.
  {
    const long long n = (long long)out_size;
    const int blocks = (int)((n + 255) / 256);
    zero_out<<<blocks, 256, 0, stream>>>(out, n);
  }

  // 3) Weighted gather + scatter-add over edges.
  {
    const long long total = (long long)E * OUT_DIM;
    const int blocks = (int)((total + 255) / 256);
    edge_scatter<<<blocks, 256, 0, stream>>>(h, ei, ei + E, ew, out, E, N, B);
  }
}